// DeformableCurrents_33225867002543
// MI455X (gfx1250) — compile-verified
//
#include <hip/hip_runtime.h>
#include <hip/hip_bf16.h>

typedef float v2f __attribute__((ext_vector_type(2)));
typedef float v8f __attribute__((ext_vector_type(8)));

#define VNUM 4096
#define NSRC 8192
#define MTAR 8192
#define PTOT 16384          // NSRC + MTAR combined point set
#define TB   1024           // PTOT / 16 tile blocks per dim

// Workspace layout: 12 component planes of PTOT floats each.
//  plane 0..2 : cx, cy, cz        (centers)
//  plane 3    : |c|^2
//  plane 4..6 : nx, ny, nz        (signed normals: -tar_normals for targets)
//  plane 7    : zeros             (K=3 pad for normal fragments)
//  plane 8..10: -2cx, -2cy, -2cz  (B-side center components)
//  plane 11   : ones              (K=3 B-side center component)

__global__ void dc_prep(const float* __restrict__ verts,
                        const float* __restrict__ tnorm,
                        const float* __restrict__ tctr,
                        const int*   __restrict__ tidx,
                        float* __restrict__ F,
                        float* __restrict__ out) {
    int g = blockIdx.x * blockDim.x + threadIdx.x;
    if (g == 0) out[0] = 0.0f;
    if (g >= PTOT) return;

    float cx, cy, cz, nx, ny, nz;
    if (g < NSRC) {
        int i0 = tidx[3 * g + 0];
        int i1 = tidx[3 * g + 1];
        int i2 = tidx[3 * g + 2];
        float ax = verts[3 * i0 + 0], ay = verts[3 * i0 + 1], az = verts[3 * i0 + 2];
        float bx = verts[3 * i1 + 0], by = verts[3 * i1 + 1], bz = verts[3 * i1 + 2];
        float ccx = verts[3 * i2 + 0], ccy = verts[3 * i2 + 1], ccz = verts[3 * i2 + 2];
        float e1x = ax - bx, e1y = ay - by, e1z = az - bz;
        float e2x = ccx - bx, e2y = ccy - by, e2z = ccz - bz;
        nx = 0.5f * (e1y * e2z - e1z * e2y);
        ny = 0.5f * (e1z * e2x - e1x * e2z);
        nz = 0.5f * (e1x * e2y - e1y * e2x);
        const float third = 1.0f / 3.0f;
        cx = (ax + bx + ccx) * third;
        cy = (ay + by + ccy) * third;
        cz = (az + bz + ccz) * third;
    } else {
        int j = g - NSRC;
        cx = tctr[3 * j + 0];
        cy = tctr[3 * j + 1];
        cz = tctr[3 * j + 2];
        nx = -tnorm[3 * j + 0];
        ny = -tnorm[3 * j + 1];
        nz = -tnorm[3 * j + 2];
    }
    float csq = cx * cx + cy * cy + cz * cz;

    F[0 * PTOT + g] = cx;
    F[1 * PTOT + g] = cy;
    F[2 * PTOT + g] = cz;
    F[3 * PTOT + g] = csq;
    F[4 * PTOT + g] = nx;
    F[5 * PTOT + g] = ny;
    F[6 * PTOT + g] = nz;
    F[7 * PTOT + g] = 0.0f;
    F[8 * PTOT + g] = -2.0f * cx;
    F[9 * PTOT + g] = -2.0f * cy;
    F[10 * PTOT + g] = -2.0f * cz;
    F[11 * PTOT + g] = 1.0f;
}

// Main kernel: one wave per (row-pair, column-stride) slot.
// grid = 512 blocks x 256 threads = 4096 waves; rpair = wave>>3, cstride = wave&7.
// Row-pair {r, 1023-r} gives every wave-slot exactly 1025/8 tiles -> perfect balance.
__global__ void __launch_bounds__(256)
dc_main(const float* __restrict__ F, float* __restrict__ out) {
    const int lane = threadIdx.x & 31;

    // Wave-uniform scalars: force into SGPRs so loop control is pure SALU and
    // EXEC provably stays all-1s across every WMMA (ISA requirement).
    const int gwave = __builtin_amdgcn_readfirstlane(blockIdx.x * 8 + (threadIdx.x >> 5));
    const int rpair = gwave >> 3;   // 0..511
    const int cstr  = gwave & 7;    // 0..7

    const int m  = lane & 15;              // row/col within tile (M for A, N for B/C)
    const int kc = (lane >> 4) << 1;       // K base: 0 for lanes 0-15, 2 for lanes 16-31

    float accDiag = 0.0f;   // diagonal tiles (weight 1)
    float accOff  = 0.0f;   // strictly-upper tiles (weight 2 via symmetry)
    v8f zero8 = {};

    for (int s = 0; s < 2; ++s) {
        const int ti = (s == 0) ? rpair : ((TB - 1) - rpair);
        const int i  = ti * 16 + m;

        // A fragments (reused for the whole column loop of this row block)
        v2f aC, aN;
        aC.x = F[(0 + kc) * PTOT + i];      // cx or cz
        aC.y = F[(1 + kc) * PTOT + i];      // cy or |c|^2
        aN.x = F[(4 + kc) * PTOT + i];      // nx or nz
        aN.y = F[(5 + kc) * PTOT + i];      // ny or 0

        auto tileSum = [&](int tj) -> float {
            const int j = tj * 16 + m;

            v2f bC, bN;
            bC.x = F[(8 + kc) * PTOT + j];  // -2cx or -2cz
            bC.y = F[(9 + kc) * PTOT + j];  // -2cy or 1
            bN.x = F[(4 + kc) * PTOT + j];  // nx or nz
            bN.y = F[(5 + kc) * PTOT + j];  // ny or 0

            // |c_j|^2 + 1 : column index N = lane&15 is fixed per lane in C/D layout
            const float csqc1 = F[3 * PTOT + j] + 1.0f;

            // d2raw[r][c] = -2 c_i.c_j + |c_i|^2   (K=4 f32 WMMA)
            v8f d2 = __builtin_amdgcn_wmma_f32_16x16x4_f32(
                false, aC, false, bC, (short)0, zero8, false, false);
            // ndot[r][c] = n_i . n_j
            v8f nd = __builtin_amdgcn_wmma_f32_16x16x4_f32(
                false, aN, false, bN, (short)0, zero8, false, false);

            float ts = 0.0f;
#pragma unroll
            for (int e = 0; e < 8; ++e) {
                // K = 1 / (1 + max(d2, 0)) = 1 / max(d2 + 1, 1)
                float denom = fmaxf(d2[e] + csqc1, 1.0f);
                ts = fmaf(nd[e], __builtin_amdgcn_rcpf(denom), ts);
            }
            return ts;
        };

        int tj = ti + cstr;
        if (cstr == 0) {                    // peel the (only possible) diagonal tile
            accDiag += tileSum(ti);
            tj = ti + 8;
        }
        for (; tj < TB; tj += 8)
            accOff += tileSum(tj);
    }

    float acc = fmaf(2.0f, accOff, accDiag);

    // wave32 reduction
#pragma unroll
    for (int off = 16; off > 0; off >>= 1)
        acc += __shfl_xor(acc, off, 32);
    if (lane == 0)
        atomicAdd(out, acc);
}

extern "C" void kernel_launch(void* const* d_in, const int* in_sizes, int n_in,
                              void* d_out, int out_size, void* d_ws, size_t ws_size,
                              hipStream_t stream) {
    const float* verts = (const float*)d_in[0];   // src_vertices  (V,3) f32
    const float* tnorm = (const float*)d_in[1];   // tar_normals   (M,3) f32
    const float* tctr  = (const float*)d_in[2];   // tar_centers   (M,3) f32
    const int*   tidx  = (const int*)  d_in[3];   // src_indices   (N,3) i32
    float* out = (float*)d_out;
    float* F   = (float*)d_ws;                    // 12 * 16384 * 4B = 768 KB

    dc_prep<<<(PTOT + 255) / 256, 256, 0, stream>>>(verts, tnorm, tctr, tidx, F, out);
    dc_main<<<512, 256, 0, stream>>>(F, out);
}